// Dynamic_Masking_Layer_68899865362757
// MI455X (gfx1250) — compile-verified
//
#include <hip/hip_runtime.h>

typedef float v2f __attribute__((ext_vector_type(2)));
typedef float v4f __attribute__((ext_vector_type(4)));
typedef float v8f __attribute__((ext_vector_type(8)));

#define BLK_M 128
#define BLK_N 256
#define BLK_K 16
#define STRIDE_A 20     // BLK_K + 4: rows 16B-aligned, conflict-free b64 A-frag reads
#define PSTRIDE 544     // floats per k-pair row: 2*256 + 32 (pair vs pair+1 rows 32 banks apart)
#define ASZ (BLK_M * STRIDE_A)        // 2560 floats per A buffer
#define BSZ ((BLK_K / 2) * PSTRIDE)   // 4352 floats per B buffer

__global__ __launch_bounds__(256)
void Dynamic_Masking_Layer_gemm_wmma_f32(const float* __restrict__ X,
                                         const float* __restrict__ Mk,
                                         const float* __restrict__ W,
                                         const float* __restrict__ Bias,
                                         float* __restrict__ Out,
                                         int M, int N, int K)
{
    // Double-buffered. B stored k-pair interleaved: (k,n) -> (k/2)*PSTRIDE + 2n + (k&1).
    __shared__ float As[2 * ASZ];
    __shared__ float Bs[2 * BSZ];

    const int tid  = threadIdx.x;
    const int lane = tid & 31;
    const int wave = tid >> 5;
    const int wm   = wave & 1;   // 2 waves along M (64 rows each)
    const int wn   = wave >> 1;  // 4 waves along N (64 cols each)

    const int m_blk = blockIdx.y * BLK_M;
    const int n_blk = blockIdx.x * BLK_N;

    // A tile 128x16: 4 float4/row -> rowA = tid/4 (0..63), 2 passes of 64 rows.
    const int rowA  = tid >> 2;
    const int col4A = tid & 3;
    // B tile 16x256 as 8 k-pairs x 64 n-quads = 512 items, 2 per thread:
    //   item i: p = tid/64 + 4*i (0..7), q = tid%64 (n-quad).
    const int pB = tid >> 6;   // 0..3
    const int qB = tid & 63;   // 0..63

    v8f acc[4][4];
#pragma unroll
    for (int im = 0; im < 4; ++im)
#pragma unroll
        for (int in = 0; in < 4; ++in)
#pragma unroll
            for (int v = 0; v < 8; ++v)
                acc[im][in][v] = 0.0f;

    const int KT = K / BLK_K;

    v4f ra[2], rw0[2], rw1[2], rm0[2], rm1[2];
    // Prologue: prefetch tile 0 into registers.
#pragma unroll
    for (int i = 0; i < 2; ++i)
        ra[i] = *(const v4f*)(X + (size_t)(m_blk + rowA + 64 * i) * K + (col4A * 4));
#pragma unroll
    for (int i = 0; i < 2; ++i) {
        const int p = pB + 4 * i;
        rw0[i] = *(const v4f*)(W  + (size_t)(2 * p + 0) * N + (n_blk + qB * 4));
        rw1[i] = *(const v4f*)(W  + (size_t)(2 * p + 1) * N + (n_blk + qB * 4));
        rm0[i] = *(const v4f*)(Mk + (size_t)(2 * p + 0) * N + (n_blk + qB * 4));
        rm1[i] = *(const v4f*)(Mk + (size_t)(2 * p + 1) * N + (n_blk + qB * 4));
    }

    for (int kt = 0; kt < KT; ++kt) {
        const int buf = kt & 1;
        float* __restrict__ as_w = As + buf * ASZ;
        float* __restrict__ bs_w = Bs + buf * BSZ;

        // Registers -> LDS; mask fused; B written as interleaved {row 2p, row 2p+1} pairs.
#pragma unroll
        for (int i = 0; i < 2; ++i)
            *(v4f*)(as_w + (rowA + 64 * i) * STRIDE_A + col4A * 4) = ra[i];
#pragma unroll
        for (int i = 0; i < 2; ++i) {
            const int p = pB + 4 * i;
#pragma unroll
            for (int j = 0; j < 4; ++j) {
                v2f pr;
                pr.x = (rm0[i][j] != 0.0f) ? rw0[i][j] : 0.0f;
                pr.y = (rm1[i][j] != 0.0f) ? rw1[i][j] : 0.0f;
                *(v2f*)(bs_w + p * PSTRIDE + (qB * 4 + j) * 2) = pr;
            }
        }
        // Single barrier per tile (double-buffered; see round-3 ordering argument).
        __syncthreads();

        // Prefetch next K-tile while we compute on this one.
        if (kt + 1 < KT) {
            const int kn = (kt + 1) * BLK_K;
#pragma unroll
            for (int i = 0; i < 2; ++i)
                ra[i] = *(const v4f*)(X + (size_t)(m_blk + rowA + 64 * i) * K + (kn + col4A * 4));
#pragma unroll
            for (int i = 0; i < 2; ++i) {
                const int p = pB + 4 * i;
                rw0[i] = *(const v4f*)(W  + (size_t)(kn + 2 * p + 0) * N + (n_blk + qB * 4));
                rw1[i] = *(const v4f*)(W  + (size_t)(kn + 2 * p + 1) * N + (n_blk + qB * 4));
                rm0[i] = *(const v4f*)(Mk + (size_t)(kn + 2 * p + 0) * N + (n_blk + qB * 4));
                rm1[i] = *(const v4f*)(Mk + (size_t)(kn + 2 * p + 1) * N + (n_blk + qB * 4));
            }
        }

        const float* __restrict__ as_r = As + buf * ASZ;
        const float* __restrict__ bs_r = Bs + buf * BSZ;

        // 4 k-steps of 16 WMMAs (4 M-subtiles x 4 N-subtiles).
#pragma unroll
        for (int s = 0; s < 4; ++s) {
            const int krow = 4 * s + 2 * (lane >> 4);   // always even
            const int prow = 2 * s + (lane >> 4);       // k-pair row index

            v2f af[4];
#pragma unroll
            for (int im = 0; im < 4; ++im) {
                const int m_l = wm * 64 + im * 16 + (lane & 15);
                af[im] = *(const v2f*)(as_r + m_l * STRIDE_A + krow);
            }

            v2f bf[4];
#pragma unroll
            for (int in = 0; in < 4; ++in) {
                const int n_l = wn * 64 + in * 16 + (lane & 15);
                bf[in] = *(const v2f*)(bs_r + prow * PSTRIDE + n_l * 2);
            }

#pragma unroll
            for (int im = 0; im < 4; ++im)
#pragma unroll
                for (int in = 0; in < 4; ++in)
                    acc[im][in] = __builtin_amdgcn_wmma_f32_16x16x4_f32(
                        false, af[im], false, bf[in],
                        (short)0, acc[im][in], false, false);
        }
    }

    // Epilogue: bias add + store. C layout: M = v + 8*(lane/16), N = lane%16.
#pragma unroll
    for (int in = 0; in < 4; ++in) {
        const int n_g = n_blk + wn * 64 + in * 16 + (lane & 15);
        const float bv = Bias[n_g];
#pragma unroll
        for (int im = 0; im < 4; ++im) {
            const int m_base = m_blk + wm * 64 + im * 16 + 8 * (lane >> 4);
#pragma unroll
            for (int v = 0; v < 8; ++v)
                Out[(size_t)(m_base + v) * N + n_g] = acc[im][in][v] + bv;
        }
    }
}

extern "C" void kernel_launch(void* const* d_in, const int* in_sizes, int n_in,
                              void* d_out, int out_size, void* d_ws, size_t ws_size,
                              hipStream_t stream) {
    const float* X    = (const float*)d_in[0];
    const float* Mask = (const float*)d_in[1];
    const float* W    = (const float*)d_in[2];
    const float* Bias = (const float*)d_in[3];
    float* Out        = (float*)d_out;

    const int N = in_sizes[3];           // 4096
    const int K = in_sizes[1] / N;       // 4096
    const int M = in_sizes[0] / K;       // 2048

    dim3 grid(N / BLK_N, M / BLK_M);
    Dynamic_Masking_Layer_gemm_wmma_f32<<<grid, dim3(256), 0, stream>>>(
        X, Mask, W, Bias, Out, M, N, K);
}